// Attention_79010218377704
// MI455X (gfx1250) — compile-verified
//
#include <hip/hip_runtime.h>

#define B_ 32
#define S_ 2048
#define D_ 1024
#define H_ 1024
#define NEGV (-1000000000000.0f)

typedef float        v2f __attribute__((ext_vector_type(2)));
typedef float        v8f __attribute__((ext_vector_type(8)));
typedef int          v4i __attribute__((ext_vector_type(4)));
typedef int          v8i __attribute__((ext_vector_type(8)));
typedef unsigned int v4u __attribute__((ext_vector_type(4)));

typedef __attribute__((address_space(1))) v4i*   g_v4i_ptr;  // global int4*
typedef __attribute__((address_space(3))) v4i*   l_v4i_ptr;  // LDS int4*
typedef __attribute__((address_space(3))) float* l_f32_ptr;  // LDS float*

static constexpr int LDS_STRIDE = 1028;  // 1024 + 4 pad floats: rows shift 4 banks -> conflict-free b64 reads
static constexpr int ROWS = 64;          // s-rows per workgroup

#if defined(__HIP_DEVICE_COMPILE__) && __has_builtin(__builtin_amdgcn_tensor_load_to_lds) && \
    __has_builtin(__builtin_amdgcn_s_wait_tensorcnt)
#define USE_TDM 1
#else
#define USE_TDM 0
#endif
#if defined(__HIP_DEVICE_COMPILE__) && __has_builtin(__builtin_amdgcn_global_load_async_to_lds_b128)
#define USE_ASYNC_LDS 1
#else
#define USE_ASYNC_LDS 0
#endif

// Hardware tanh: V_TANH_F32 (CDNA5 trans op) instead of the ~40-op libcall expansion.
__device__ __forceinline__ float fast_tanh(float x) {
#if defined(__HIP_DEVICE_COMPILE__) && __has_builtin(__builtin_amdgcn_tanhf)
  return __builtin_amdgcn_tanhf(x);
#elif defined(__HIP_DEVICE_COMPILE__)
  float y;
  asm volatile("v_tanh_f32 %0, %1\n\tv_nop" : "=v"(y) : "v"(x));  // v_nop covers TRANS RAW hazard
  return y;
#else
  return tanhf(x);
#endif
}

// ---------------- Kernel A: hb[b][h] = bias[h] + dot(last_state[b], Wh[h]) ----------------
__global__ void hpart_kernel(const float* __restrict__ ls,
                             const float* __restrict__ W,
                             const float* __restrict__ bias,
                             float* __restrict__ hb) {
  __shared__ float lsb[D_];
  const int b = blockIdx.y;
  const int h = blockIdx.x * blockDim.x + threadIdx.x;
  for (int d = threadIdx.x; d < D_; d += blockDim.x) lsb[d] = ls[(size_t)b * D_ + d];
  __syncthreads();
  const float* wrow = W + (size_t)h * (2 * D_);   // Wh = W[h][0:D]
  float acc = 0.f;
#pragma unroll 4
  for (int d = 0; d < D_; d += 4) {
    float4 w4 = *(const float4*)(wrow + d);
    acc += lsb[d] * w4.x + lsb[d + 1] * w4.y + lsb[d + 2] * w4.z + lsb[d + 3] * w4.w;
  }
  hb[(size_t)b * H_ + h] = acc + bias[h];
}

// ---------------- Kernel B: fused  e_part GEMM (WMMA f32) + tanh + V-dot + mask ----------------
// grid (S/ROWS, B), block 256 (8 waves). Wave w: s_subtile = w>>1, h_half = w&1.
__global__ void scores_kernel(const float* __restrict__ enc,
                              const float* __restrict__ W,
                              const float* __restrict__ V,
                              const float* __restrict__ hb,
                              const int*  __restrict__ mask,
                              float* __restrict__ scores_out) {
  extern __shared__ float smem[];
  float* encT  = smem;                       // ROWS x LDS_STRIDE enc tile
  float* scAcc = smem + ROWS * LDS_STRIDE;   // ROWS partial scores

  const int b      = blockIdx.y;
  const int s_base = blockIdx.x * ROWS;
  const int tid    = threadIdx.x;
  const int wave   = tid >> 5;

  // Stage enc[b, s_base:s_base+64, :] into LDS.
  const float* encB = enc + ((size_t)b * S_ + s_base) * D_;
#if USE_TDM
  // Tensor Data Mover: ONE instruction moves the whole 64x1024 f32 tile, using the
  // D# "iterate" feature to write rows at an LDS stride of 1028 floats (our padded layout).
  if (wave == 0) {
    const unsigned long long ga = (unsigned long long)(uintptr_t)encB;   // byte address
    const unsigned lds_base = (unsigned)(uintptr_t)(l_f32_ptr)encT;     // LDS byte offset
    // Group0: count=1 | lds_addr | global_addr[56:0] | type=2
    v4u g0 = { 1u,
               lds_base,
               (unsigned)(ga & 0xFFFFFFFFu),
               (unsigned)((ga >> 32) & 0x01FFFFFFu) | 0x80000000u };
    // Group1: data_size=4B (2<<16) | iterate_enable (1<<19); tensor_dim0=1024; tensor_dim1=1;
    //         tile_dim0=1024; tile_dim1=1; tensor_dim0_stride=1024
    v8i g1 = { 0x000A0000,           // mask=0 | data_size=2 | iterate_enable
               0x04000000,           // abar_addr=0 | tensor_dim0.lo16=1024
               0x00010000,           // tensor_dim0.hi16=0 | tensor_dim1.lo16=1
               0x04000000,           // tensor_dim1.hi16=0 | tile_dim0=1024
               1,                    // tile_dim1=1 | tile_dim2=0
               1024, 0, 0 };         // tensor_dim0_stride=1024; dim1_stride=0
    // Group2 (iterate mode): lds_addr_increment=1028 elems, global_addr_increment=1024 elems,
    //                        iterate_count=63 -> 64 rows
    v4i g2 = { 0, 1028, 1024, (63 << 16) };
    v4i g3 = { 0, 0, 0, 0 };
    v8i g4 = { 0, 0, 0, 0, 0, 0, 0, 0 };   // unused extra group (6-arg builtin form)
    __builtin_amdgcn_tensor_load_to_lds(g0, g1, g2, g3, g4, 0);
    __builtin_amdgcn_s_wait_tensorcnt(0);
  }
#elif USE_ASYNC_LDS
  for (int idx = tid; idx < ROWS * (D_ / 4); idx += 256) {
    const int row = idx >> 8;        // D_/4 == 256 float4 per row
    const int c4  = idx & 255;
    __builtin_amdgcn_global_load_async_to_lds_b128(
        (g_v4i_ptr)(encB + (size_t)row * D_ + c4 * 4),
        (l_v4i_ptr)(encT + row * LDS_STRIDE + c4 * 4),
        0, 0);
  }
  asm volatile("s_wait_asynccnt 0x0" ::: "memory");
#else
  for (int idx = tid; idx < ROWS * (D_ / 4); idx += 256) {
    const int row = idx >> 8;
    const int c4  = idx & 255;
    float4 v4 = *(const float4*)(encB + (size_t)row * D_ + c4 * 4);
    *(float4*)(encT + row * LDS_STRIDE + c4 * 4) = v4;
  }
#endif
  if (tid < ROWS) scAcc[tid] = 0.f;
  __syncthreads();

  const int lane   = tid & 31;
  const int s_sub  = wave >> 1;
  const int h_half = wave & 1;
  const int n      = lane & 15;   // N index (h within tile) / M index of A row
  const int hi     = lane >> 4;   // selects K+2,K+3 half per WMMA fragment layout

  // A fragment source: row (s_sub*16 + n), K offset hi*2  (ds_load_b64 per step)
  const float* aBase = encT + (s_sub * 16 + n) * LDS_STRIDE + hi * 2;

  float p[8];
#pragma unroll
  for (int r = 0; r < 8; ++r) p[r] = 0.f;

  for (int t = 0; t < H_ / 32; ++t) {          // 32 h-tiles of 16 per wave
    const int h_base = h_half * (H_ / 2) + t * 16;
    const float hbv = hb[(size_t)b * H_ + h_base + n];  // C layout: N = lane%16
    const float vv  = V[h_base + n];
    // B fragment: B[k][n] = We[n][k] = W[h][D_ + k]
    const float* wrow = W + (size_t)(h_base + n) * (2 * D_) + D_ + hi * 2;

    v8f c0, c1;
#pragma unroll
    for (int r = 0; r < 8; ++r) { c0[r] = hbv; c1[r] = 0.f; }

#pragma unroll 4
    for (int k = 0; k < D_; k += 8) {          // two accumulators break the WMMA RAW chain
      v2f a0 = *(const v2f*)(aBase + k);
      v2f b0 = *(const v2f*)(wrow + k);
      c0 = __builtin_amdgcn_wmma_f32_16x16x4_f32(false, a0, false, b0, (short)0, c0, false, false);
      v2f a1 = *(const v2f*)(aBase + k + 4);
      v2f b1 = *(const v2f*)(wrow + k + 4);
      c1 = __builtin_amdgcn_wmma_f32_16x16x4_f32(false, a1, false, b1, (short)0, c1, false, false);
    }
#pragma unroll
    for (int r = 0; r < 8; ++r) p[r] += vv * fast_tanh(c0[r] + c1[r]);
  }

  // Reduce over N (16 lanes) — masks < 16 keep the two C-halves separate.
#pragma unroll
  for (int r = 0; r < 8; ++r) {
    float x = p[r];
    x += __shfl_xor(x, 8, 32);
    x += __shfl_xor(x, 4, 32);
    x += __shfl_xor(x, 2, 32);
    x += __shfl_xor(x, 1, 32);
    p[r] = x;
  }
  if (n == 0) {                                // lanes 0 (M=r) and 16 (M=r+8)
#pragma unroll
    for (int r = 0; r < 8; ++r)
      atomicAdd(&scAcc[s_sub * 16 + hi * 8 + r], p[r]);  // exactly 2 adds/slot: deterministic
  }
  __syncthreads();

  if (tid < ROWS) {
    const int s = s_base + tid;
    float sc = scAcc[tid];
    if (mask[(size_t)b * S_ + s] != 1) sc = NEGV;
    scores_out[(size_t)b * S_ + s] = sc;       // reference returns the MASKED scores
  }
}

// ---------------- Kernel C: per-batch softmax over S ----------------
__global__ void softmax_kernel(const float* __restrict__ scores,
                               float* __restrict__ attn) {
  __shared__ float red[256];
  const int b = blockIdx.x;
  const int tid = threadIdx.x;
  const float* sc = scores + (size_t)b * S_;
  float mx = -3.402823466e38f;
  for (int s = tid; s < S_; s += 256) mx = fmaxf(mx, sc[s]);
  red[tid] = mx; __syncthreads();
  for (int off = 128; off > 0; off >>= 1) {
    if (tid < off) red[tid] = fmaxf(red[tid], red[tid + off]);
    __syncthreads();
  }
  mx = red[0]; __syncthreads();
  float sum = 0.f;
  for (int s = tid; s < S_; s += 256) sum += expf(sc[s] - mx);
  red[tid] = sum; __syncthreads();
  for (int off = 128; off > 0; off >>= 1) {
    if (tid < off) red[tid] += red[tid + off];
    __syncthreads();
  }
  const float inv = 1.f / red[0];
  for (int s = tid; s < S_; s += 256) attn[(size_t)b * S_ + s] = expf(sc[s] - mx) * inv;
}

// ---------------- Kernel D1/D2: context = attn @ enc (two-stage, deterministic) ----------------
__global__ void context_partial_kernel(const float* __restrict__ attn,
                                       const float* __restrict__ enc,
                                       float* __restrict__ part) {
  const int dcol = blockIdx.x * 256 + threadIdx.x;
  const int sch  = blockIdx.y;             // 8 chunks of 256 s-values
  const int b    = blockIdx.z;
  const float* encB = enc + ((size_t)b * S_ + sch * 256) * D_ + dcol;
  const float* w = attn + (size_t)b * S_ + sch * 256;
  float acc = 0.f;
#pragma unroll 4
  for (int i = 0; i < 256; ++i) acc += w[i] * encB[(size_t)i * D_];
  part[((size_t)b * 8 + sch) * D_ + dcol] = acc;
}

__global__ void context_reduce_kernel(const float* __restrict__ part,
                                      float* __restrict__ ctx) {
  const int dcol = blockIdx.x * 256 + threadIdx.x;
  const int b    = blockIdx.y;
  float acc = 0.f;
#pragma unroll
  for (int j = 0; j < 8; ++j) acc += part[((size_t)b * 8 + j) * D_ + dcol];
  ctx[(size_t)b * D_ + dcol] = acc;
}

// ---------------- launch ----------------
extern "C" void kernel_launch(void* const* d_in, const int* in_sizes, int n_in,
                              void* d_out, int out_size, void* d_ws, size_t ws_size,
                              hipStream_t stream) {
  const float* last_state = (const float*)d_in[0];
  const float* enc        = (const float*)d_in[1];
  const int*   mask       = (const int*)  d_in[2];
  const float* W          = (const float*)d_in[3];
  const float* bias       = (const float*)d_in[4];
  const float* V          = (const float*)d_in[5];

  float* out    = (float*)d_out;
  float* ctx    = out;                         // B*D
  float* attn   = out + B_ * D_;               // B*S
  float* scores = out + B_ * D_ + B_ * S_;     // B*S (masked)

  float* hb   = (float*)d_ws;                  // B*H
  float* part = hb + B_ * H_;                  // B*8*D

  hpart_kernel<<<dim3(H_ / 256, B_), 256, 0, stream>>>(last_state, W, bias, hb);

  const size_t smem = (size_t)(ROWS * LDS_STRIDE + ROWS) * sizeof(float);  // ~257 KB < 320 KB/WGP
  scores_kernel<<<dim3(S_ / ROWS, B_), 256, smem, stream>>>(enc, W, V, hb, mask, scores);

  softmax_kernel<<<B_, 256, 0, stream>>>(scores, attn);

  context_partial_kernel<<<dim3(D_ / 256, 8, B_), 256, 0, stream>>>(attn, enc, part);
  context_reduce_kernel<<<dim3(D_ / 256, B_), 256, 0, stream>>>(part, ctx);
}